// AUGRU_17257178596030
// MI455X (gfx1250) — compile-verified
//
#include <hip/hip_runtime.h>

typedef __attribute__((ext_vector_type(2))) float v2f;
typedef __attribute__((ext_vector_type(8))) float v8f;

#define WMMA_F32(a, b, c) \
  __builtin_amdgcn_wmma_f32_16x16x4_f32(false, (a), false, (b), (short)0, (c), false, false)

constexpr int Bc = 1024;   // batch
constexpr int Tc = 200;    // time steps
constexpr int Ic = 128;    // input dim
constexpr int Hc = 128;    // hidden dim
constexpr int PITCH = 132; // LDS row pitch in floats (64-bank conflict-free WMMA operand reads)
constexpr int BT = 32;     // batch tile per scan block

// ---------------------------------------------------------------------------
// Phase 1: xg[t][b][:] = inputs[b][t][:] @ Wg_x^T + bg   for g in {z,r,h}
// Output layout (T, B, H) fp32. One wave = one 16-row M tile, full N=128.
// Inner loop is software-pipelined: B tiles for k+1 are loaded into distinct
// registers while the 8 WMMAs for k issue.
// ---------------------------------------------------------------------------
__global__ __launch_bounds__(256) void augru_proj(
    const float* __restrict__ x,  // (B,T,I) flat; row r=b*T+t at r*I
    const float* __restrict__ Wz, const float* __restrict__ bz,
    const float* __restrict__ Wr, const float* __restrict__ br,
    const float* __restrict__ Wh, const float* __restrict__ bh,
    float* __restrict__ xz, float* __restrict__ xr, float* __restrict__ xh)
{
  extern __shared__ float lds[];  // Hc * PITCH floats: W_g_x as [n][k]

  const int gate = blockIdx.y;
  const float* W    = (gate == 0) ? Wz : (gate == 1) ? Wr : Wh;
  const float* bias = (gate == 0) ? bz : (gate == 1) ? br : bh;
  float* outp       = (gate == 0) ? xz : (gate == 1) ? xr : xh;

  const int tid = threadIdx.x;

  // Stage W_g_x (first Ic columns of (Hc, Ic+Hc)) into LDS, row-major [n][k].
  for (int idx = tid; idx < Hc * Ic; idx += 256) {
    int n = idx >> 7, k = idx & 127;
    lds[n * PITCH + k] = W[n * (Ic + Hc) + k];
  }
  __syncthreads();

  const int wave = tid >> 5, lane = tid & 31;
  const int lmod = lane & 15, ldiv = lane >> 4;
  const long mt = (long)blockIdx.x * 8 + wave;  // M-tile index (16 rows)
  const long rbase = mt * 16;

  // bias per N column (independent of M row)
  float biasv[8];
#pragma unroll
  for (int nt = 0; nt < 8; ++nt) biasv[nt] = bias[nt * 16 + lmod];

  v8f acc[8] = {};

  // A operand: lane holds row M=lmod, K = kk + 2*ldiv + {0,1}
  const float* arow = x + (rbase + lmod) * Ic + 2 * ldiv;
  const float* brow = lds + lmod * PITCH + 2 * ldiv;  // + nt*16*PITCH + kk

  // ---- software-pipelined K loop -------------------------------------------
  v2f a_c = *(const v2f*)(arow);
  v2f b_c[8];
#pragma unroll
  for (int nt = 0; nt < 8; ++nt) b_c[nt] = *(const v2f*)(brow + nt * 16 * PITCH);

#pragma unroll 4
  for (int kk = 0; kk < Ic - 4; kk += 4) {
    v2f a_n = *(const v2f*)(arow + kk + 4);
    v2f b_n[8];
#pragma unroll
    for (int nt = 0; nt < 8; ++nt)
      b_n[nt] = *(const v2f*)(brow + nt * 16 * PITCH + kk + 4);
#pragma unroll
    for (int nt = 0; nt < 8; ++nt) acc[nt] = WMMA_F32(a_c, b_c[nt], acc[nt]);
    a_c = a_n;
#pragma unroll
    for (int nt = 0; nt < 8; ++nt) b_c[nt] = b_n[nt];
  }
#pragma unroll
  for (int nt = 0; nt < 8; ++nt) acc[nt] = WMMA_F32(a_c, b_c[nt], acc[nt]);

  // Store: C vgpr v holds rows M=v and M=v+8 (lanes 16-31), N = nt*16+lmod.
#pragma unroll
  for (int v = 0; v < 8; ++v) {
    long r = rbase + v + 8 * ldiv;
    int  tt = (int)(r % Tc);
    long bb = r / Tc;
    float* orow = outp + ((long)tt * Bc + bb) * Hc + lmod;
#pragma unroll
    for (int nt = 0; nt < 8; ++nt) orow[nt * 16] = acc[nt][v] + biasv[nt];
  }
}

// ---------------------------------------------------------------------------
// Phase 2: 200-step AUGRU scan. One block owns 32 batch rows; recurrent
// weights + h-state + r*h staging live in LDS (~237 KB of the 320 KB WGP LDS).
// x-preactivations are loaded into registers at step start (overlapping the
// GEMMs) and added at activation time; GEMM operand fetch is pipelined.
// ---------------------------------------------------------------------------
__global__ __launch_bounds__(256) void augru_scan(
    const float* __restrict__ att,  // (B,T)
    const float* __restrict__ Wz, const float* __restrict__ Wr,
    const float* __restrict__ Wh,
    const float* __restrict__ xz, const float* __restrict__ xr,
    const float* __restrict__ xh,  // each (T,B,H)
    float* __restrict__ out)       // (B,H)
{
  extern __shared__ float lds[];
  float* wzl   = lds;                    // Hc*PITCH: Wz_h as [n][k]
  float* wrl   = wzl + Hc * PITCH;
  float* whl   = wrl + Hc * PITCH;
  float* hbuf  = whl + Hc * PITCH;       // BT*PITCH: h state [m][n]
  float* rhbuf = hbuf + BT * PITCH;      // BT*PITCH: r ⊙ h
  float* abuf  = rhbuf + BT * PITCH;     // BT: a_t per batch row

  const int tid = threadIdx.x;
  const int b0  = blockIdx.x * BT;

  // Stage recurrent halves (columns Ic..Ic+Hc-1) of each weight matrix; zero h.
  for (int idx = tid; idx < Hc * Hc; idx += 256) {
    int n = idx >> 7, k = idx & 127;
    wzl[n * PITCH + k] = Wz[n * (Ic + Hc) + Ic + k];
    wrl[n * PITCH + k] = Wr[n * (Ic + Hc) + Ic + k];
    whl[n * PITCH + k] = Wh[n * (Ic + Hc) + Ic + k];
  }
  for (int idx = tid; idx < BT * Hc; idx += 256)
    hbuf[(idx >> 7) * PITCH + (idx & 127)] = 0.f;
  __syncthreads();

  const int wave = tid >> 5, lane = tid & 31;
  const int lmod = lane & 15, ldiv = lane >> 4;
  const int nt   = wave;               // each wave owns one 16-wide N tile
  const int ncol = nt * 16 + lmod;     // N coordinate of this lane

  const float* wzr = wzl + ncol * PITCH + 2 * ldiv;
  const float* wrr = wrl + ncol * PITCH + 2 * ldiv;
  const float* whr = whl + ncol * PITCH + 2 * ldiv;
  const float* ha0 = hbuf + lmod * PITCH + 2 * ldiv;
  const float* ha1 = hbuf + (16 + lmod) * PITCH + 2 * ldiv;
  const float* ra0 = rhbuf + lmod * PITCH + 2 * ldiv;
  const float* ra1 = rhbuf + (16 + lmod) * PITCH + 2 * ldiv;

  for (int t = 0; t < Tc; ++t) {
    if (tid < BT) abuf[tid] = att[(long)(b0 + tid) * Tc + t];

    // Prefetch next step's preactivation slices (this block's 48 KB).
    if (t + 1 < Tc) {
      long nb = ((long)(t + 1) * Bc + b0) * Hc + tid * 16;
      __builtin_prefetch(xz + nb, 0, 1);
      __builtin_prefetch(xr + nb, 0, 1);
      __builtin_prefetch(xh + nb, 0, 1);
    }

    // Issue x-preactivation loads now; they complete while the GEMMs run.
    const long xbase = ((long)t * Bc + b0) * Hc;
    float xzv[16], xrv[16], xhv[16];  // [mt*8 + v]
#pragma unroll
    for (int mt = 0; mt < 2; ++mt) {
      long base = xbase + (long)(mt * 16 + 8 * ldiv) * Hc + ncol;
#pragma unroll
      for (int v = 0; v < 8; ++v) {
        xzv[mt * 8 + v] = xz[base + (long)v * Hc];
        xrv[mt * 8 + v] = xr[base + (long)v * Hc];
        xhv[mt * 8 + v] = xh[base + (long)v * Hc];
      }
    }

    v8f az[2] = {}, ar[2] = {}, ah[2] = {};

    // ---- z and r GEMMs: A = h (LDS), B = Wz_h / Wr_h (LDS), pipelined ------
    {
      v2f bz_c = *(const v2f*)(wzr);
      v2f br_c = *(const v2f*)(wrr);
      v2f a0_c = *(const v2f*)(ha0);
      v2f a1_c = *(const v2f*)(ha1);
#pragma unroll 4
      for (int kk = 0; kk < Hc - 4; kk += 4) {
        v2f bz_n = *(const v2f*)(wzr + kk + 4);
        v2f br_n = *(const v2f*)(wrr + kk + 4);
        v2f a0_n = *(const v2f*)(ha0 + kk + 4);
        v2f a1_n = *(const v2f*)(ha1 + kk + 4);
        az[0] = WMMA_F32(a0_c, bz_c, az[0]);
        az[1] = WMMA_F32(a1_c, bz_c, az[1]);
        ar[0] = WMMA_F32(a0_c, br_c, ar[0]);
        ar[1] = WMMA_F32(a1_c, br_c, ar[1]);
        bz_c = bz_n; br_c = br_n; a0_c = a0_n; a1_c = a1_n;
      }
      az[0] = WMMA_F32(a0_c, bz_c, az[0]);
      az[1] = WMMA_F32(a1_c, bz_c, az[1]);
      ar[0] = WMMA_F32(a0_c, br_c, ar[0]);
      ar[1] = WMMA_F32(a1_c, br_c, ar[1]);
    }

    // Gates: keep a*... sigmoid(z) in registers; write r ⊙ h to LDS.
#pragma unroll
    for (int mt = 0; mt < 2; ++mt) {
#pragma unroll
      for (int v = 0; v < 8; ++v) {
        int m = mt * 16 + v + 8 * ldiv;
        float zpre = xzv[mt * 8 + v] + az[mt][v];
        float rpre = xrv[mt * 8 + v] + ar[mt][v];
        az[mt][v] = 1.f / (1.f + __expf(-zpre));   // sigmoid(z)
        float rs  = 1.f / (1.f + __expf(-rpre));   // sigmoid(r)
        rhbuf[m * PITCH + ncol] = rs * hbuf[m * PITCH + ncol];
      }
    }
    __syncthreads();  // rhbuf (and abuf) visible to all waves

    // ---- h~ GEMM: A = r⊙h (LDS), B = Wh_h (LDS), pipelined -----------------
    {
      v2f b_c  = *(const v2f*)(whr);
      v2f a0_c = *(const v2f*)(ra0);
      v2f a1_c = *(const v2f*)(ra1);
#pragma unroll 4
      for (int kk = 0; kk < Hc - 4; kk += 4) {
        v2f b_n  = *(const v2f*)(whr + kk + 4);
        v2f a0_n = *(const v2f*)(ra0 + kk + 4);
        v2f a1_n = *(const v2f*)(ra1 + kk + 4);
        ah[0] = WMMA_F32(a0_c, b_c, ah[0]);
        ah[1] = WMMA_F32(a1_c, b_c, ah[1]);
        b_c = b_n; a0_c = a0_n; a1_c = a1_n;
      }
      ah[0] = WMMA_F32(a0_c, b_c, ah[0]);
      ah[1] = WMMA_F32(a1_c, b_c, ah[1]);
    }

    // h_new = (1 - a*z)*h + (a*z)*tanh(xh + gemm); each element one lane.
#pragma unroll
    for (int mt = 0; mt < 2; ++mt) {
#pragma unroll
      for (int v = 0; v < 8; ++v) {
        int m = mt * 16 + v + 8 * ldiv;
        float zp   = abuf[m] * az[mt][v];
        float ht   = tanhf(xhv[mt * 8 + v] + ah[mt][v]);
        float hold = hbuf[m * PITCH + ncol];
        hbuf[m * PITCH + ncol] = (1.f - zp) * hold + zp * ht;
      }
    }
    __syncthreads();  // h fully updated before next step's GEMMs
  }

  for (int idx = tid; idx < BT * Hc; idx += 256) {
    int m = idx >> 7, n = idx & 127;
    out[(long)(b0 + m) * Hc + n] = hbuf[m * PITCH + n];
  }
}

// ---------------------------------------------------------------------------
extern "C" void kernel_launch(void* const* d_in, const int* in_sizes, int n_in,
                              void* d_out, int out_size, void* d_ws, size_t ws_size,
                              hipStream_t stream) {
  const float* x   = (const float*)d_in[0];  // inputs (B,T,I)
  const float* att = (const float*)d_in[1];  // attention_scores (B,T)
  const float* Wz  = (const float*)d_in[2];
  const float* bz  = (const float*)d_in[3];
  const float* Wr  = (const float*)d_in[4];
  const float* br  = (const float*)d_in[5];
  const float* Wh  = (const float*)d_in[6];
  const float* bh  = (const float*)d_in[7];
  float* out = (float*)d_out;

  const size_t TBH = (size_t)Tc * Bc * Hc;
  float* xz = (float*)d_ws;       // (T,B,H) preactivation workspace
  float* xr = xz + TBH;
  float* xh = xr + TBH;

  // Phase 1: 12800 M-tiles / 8 waves per block, one gate per blockIdx.y.
  dim3 g1((Bc * Tc / 16) / 8, 3);
  size_t sh1 = (size_t)Hc * PITCH * sizeof(float);
  augru_proj<<<g1, 256, sh1, stream>>>(x, Wz, bz, Wr, br, Wh, bh, xz, xr, xh);

  // Phase 2: 32 blocks × 32 batch rows, scan over T in-kernel.
  size_t sh2 = (size_t)(3 * Hc * PITCH + 2 * BT * PITCH + BT) * sizeof(float);
  augru_scan<<<Bc / BT, 256, sh2, stream>>>(att, Wz, Wr, Wh, xz, xr, xh, out);
}